// RandomElasticDeformation3D_19971597926699
// MI455X (gfx1250) — compile-verified
//
#include <hip/hip_runtime.h>
#include <hip/hip_bf16.h>

#define ALPHA_F 35.0f
#define SIGMA_F 2.5f

constexpr int Bb = 2, Dd = 128, Hh = 160, Ww = 160;
constexpr int IMG_ELEMS = Bb * Dd * Hh * Ww * 2;   // 13,107,200

typedef __attribute__((ext_vector_type(2))) float v2f;
typedef __attribute__((ext_vector_type(8))) float v8f;

#if defined(__HIP_DEVICE_COMPILE__)
typedef __attribute__((address_space(1))) int  glb_i32;   // global (AS1) dword
typedef __attribute__((address_space(3))) int  lds_i32;   // LDS    (AS3) dword
#endif

// Keys cubic kernel, a = -0.5 (matches jax.image.resize 'cubic')
__device__ __forceinline__ float keys_cubic(float s) {
    s = fabsf(s);
    if (s <= 1.0f) return ((1.5f * s - 2.5f) * s) * s + 1.0f;
    if (s <  2.0f) return (((-0.5f * s) + 2.5f) * s - 4.0f) * s + 2.0f;
    return 0.0f;
}

// Build fused (Gaussian-smooth ∘ cubic-upsample) per-axis matrices into ws:
//   ws[   0 ..  511] : M_d  (128 x 4), scaled by ALPHA
//   ws[ 512 .. 1151] : M_h  (160 x 4)
//   ws[1152 .. 1791] : M_w  (160 x 4)
// Gaussian: 16 taps at offsets -7..+8 (tf.range(-(k//2)+1, k//2+1), k=17),
// conv 'SAME' zero-padding => pad_lo = 7, out-of-range source rows skipped.
__global__ void build_tables_kernel(float* __restrict__ M) {
    int r = threadIdx.x;
    if (r >= 448) return;
    int x, Nout, off; float scale;
    if      (r < 128) { x = r;       Nout = 128; off = 0;    scale = ALPHA_F; }
    else if (r < 288) { x = r - 128; Nout = 160; off = 512;  scale = 1.0f;   }
    else              { x = r - 288; Nout = 160; off = 1152; scale = 1.0f;   }

    float g[16]; float gs = 0.0f;
    #pragma unroll
    for (int t = 0; t < 16; ++t) {
        float a = (float)t - 7.0f;
        g[t] = __expf(-(a * a) / (2.0f * SIGMA_F * SIGMA_F));
        gs += g[t];
    }
    float ginv = 1.0f / gs;

    float acc0 = 0.f, acc1 = 0.f, acc2 = 0.f, acc3 = 0.f;
    for (int t = 0; t < 16; ++t) {
        int s = x + t - 7;
        if (s < 0 || s >= Nout) continue;               // 'SAME' zero pad
        float gt = g[t] * ginv;
        float src = (s + 0.5f) * (4.0f / (float)Nout) - 0.5f;
        float fs = floorf(src);
        float fr = src - fs;
        int   i0 = (int)fs;
        #pragma unroll
        for (int q = 0; q < 4; ++q) {
            float wq = keys_cubic(fr + 1.0f - (float)q);
            int idx = i0 - 1 + q;
            idx = idx < 0 ? 0 : (idx > 3 ? 3 : idx);    // clamp-accumulate at edges
            float val = gt * wq;
            if      (idx == 0) acc0 += val;
            else if (idx == 1) acc1 += val;
            else if (idx == 2) acc2 += val;
            else               acc3 += val;
        }
    }
    float* o = M + off + x * 4;
    o[0] = scale * acc0; o[1] = scale * acc1;
    o[2] = scale * acc2; o[3] = scale * acc3;
}

// 128 threads = 4 waves per block; wave w owns the 16x16 (h,w) tile at d = 4*dgrp + w.
// flow tile (per component) = WMMA( A = M_h_tile(16x4) @ T2(4x4),  B = M_w_tileᵀ(4x16) ).
// cf block staged Memory->LDS via GLOBAL_LOAD_ASYNC_TO_LDS (ASYNCcnt) when available.
__global__ __launch_bounds__(128) void deform_kernel(
    const float* __restrict__ img, const float* __restrict__ lbl,
    const float* __restrict__ cf,  const float* __restrict__ M,
    float* __restrict__ out_img,   float* __restrict__ out_lbl)
{
    const int b    = blockIdx.z >> 5;          // blockIdx.z = b*32 + dgrp
    const int dgrp = blockIdx.z & 31;
    const int wave = threadIdx.x >> 5;         // 0..3
    const int d    = dgrp * 4 + wave;
    const int h0   = blockIdx.y << 4;
    const int w0   = blockIdx.x << 4;
    const int l    = threadIdx.x & 31;
    const int half = l >> 4;                   // 0: lanes 0-15, 1: lanes 16-31
    const int lm   = l & 15;
    const int tid  = threadIdx.x;

    const int bD = b * Dd;

    // Warm L2 along the streaming front (gfx1250 global_prefetch_b8)
    __builtin_prefetch(img + (size_t)(((bD + d) * Hh + h0) * Ww + w0) * 2, 0, 1);
    __builtin_prefetch(lbl + (size_t)(((bD + d) * Hh + h0) * Ww + w0),     0, 1);

    // ---- stage coarse flow (this batch, 192 f32) into LDS ----
    __shared__ float s_cf[192];                // (i,j,k,c) : ((i*4+j)*4+k)*3+c
#if defined(__HIP_DEVICE_COMPILE__) && __has_builtin(__builtin_amdgcn_global_load_async_to_lds_b32)
    for (int t = tid; t < 192; t += 128) {
        __builtin_amdgcn_global_load_async_to_lds_b32(
            (glb_i32*)(cf + (size_t)b * 192 + t),    // global src (per-lane dword)
            (lds_i32*)&s_cf[t],                      // LDS dst    (per-lane dword)
            0, 0);
    }
  #if __has_builtin(__builtin_amdgcn_s_wait_asynccnt)
    __builtin_amdgcn_s_wait_asynccnt(0);
  #else
    asm volatile("s_wait_asynccnt 0x0" ::: "memory");
  #endif
    __syncthreads();
    for (int t = tid; t < 192; t += 128)             // map [0,1) -> [-1,1) in LDS
        s_cf[t] = s_cf[t] * 2.0f - 1.0f;
    __syncthreads();
#else
    for (int t = tid; t < 192; t += 128)
        s_cf[t] = cf[(size_t)b * 192 + t] * 2.0f - 1.0f;
    __syncthreads();
#endif

    // ---- per-axis fused weights ----
    const float md0 = M[d * 4 + 0], md1 = M[d * 4 + 1];
    const float md2 = M[d * 4 + 2], md3 = M[d * 4 + 3];
    const float* MhR = M + 512  + (h0 + lm) * 4;
    const float mh[4] = { MhR[0], MhR[1], MhR[2], MhR[3] };
    const float* MwR = M + 1152 + (w0 + lm) * 4;

    // f32 WMMA operand lane layout: lane holds K = 2*half + vgpr
    const int kB = half * 2;
    v2f bmat; bmat.x = MwR[kB]; bmat.y = MwR[kB + 1];   // B[k, n=lm] = M_w[w0+n, k]

    v8f flow[3];
    #pragma unroll
    for (int c = 0; c < 3; ++c) {
        float av[2];
        #pragma unroll
        for (int kk = 0; kk < 2; ++kk) {
            const int k = kB + kk;
            float s = 0.0f;
            #pragma unroll
            for (int j = 0; j < 4; ++j) {
                // T2[j,k] = sum_i M_d[d,i] * cf2[i,j,k,c]   (stride over i = 48)
                const float* p = s_cf + (j * 4 + k) * 3 + c;
                float t2 = md0 * p[0] + md1 * p[48] + md2 * p[96] + md3 * p[144];
                s = fmaf(mh[j], t2, s);                 // A[m=lm,k] += M_h[h0+m,j]*T2[j,k]
            }
            av[kk] = s;
        }
        v2f a; a.x = av[0]; a.y = av[1];
        v8f z = {};
        // D(16x16) = A(16x4) x B(4x16) + 0   -> v_wmma_f32_16x16x4_f32
        flow[c] = __builtin_amdgcn_wmma_f32_16x16x4_f32(
            /*neg_a=*/false, a, /*neg_b=*/false, bmat,
            /*c_mod=*/(short)0, z, /*reuse_a=*/false, /*reuse_b=*/false);
    }

    // C/D layout: vgpr v -> row m = v + 8*half ; col n = lm
    const int hh_base = h0 + half * 8;
    const int ww      = w0 + lm;
    const float2* im2 = (const float2*)img;

    #pragma unroll
    for (int v = 0; v < 8; ++v) {
        const int hh = hh_base + v;
        float cd = (float)d  + flow[0][v];
        float ch = (float)hh + flow[1][v];
        float cw = (float)ww + flow[2][v];

        // ---- trilinear image warp ----
        float fd = floorf(cd), fh = floorf(ch), fw = floorf(cw);
        float wd = cd - fd, wh = ch - fh, wf = cw - fw;
        int d0  = (int)fminf(fmaxf(fd,        0.f), 127.f);
        int d1  = (int)fminf(fmaxf(fd + 1.f,  0.f), 127.f);
        int h0i = (int)fminf(fmaxf(fh,        0.f), 159.f);
        int h1i = (int)fminf(fmaxf(fh + 1.f,  0.f), 159.f);
        int w0i = (int)fminf(fmaxf(fw,        0.f), 159.f);
        int w1i = (int)fminf(fmaxf(fw + 1.f,  0.f), 159.f);

        int p000 = ((bD + d0) * Hh + h0i) * Ww + w0i;
        int p100 = ((bD + d1) * Hh + h0i) * Ww + w0i;
        int p010 = ((bD + d0) * Hh + h1i) * Ww + w0i;
        int p110 = ((bD + d1) * Hh + h1i) * Ww + w0i;
        int dw   = w1i - w0i;

        float2 c000 = im2[p000], c001 = im2[p000 + dw];
        float2 c100 = im2[p100], c101 = im2[p100 + dw];
        float2 c010 = im2[p010], c011 = im2[p010 + dw];
        float2 c110 = im2[p110], c111 = im2[p110 + dw];

        #define LRP2(a, b, t) make_float2(a.x + (t) * (b.x - a.x), a.y + (t) * (b.y - a.y))
        float2 c00v = LRP2(c000, c100, wd);
        float2 c01v = LRP2(c001, c101, wd);
        float2 c10v = LRP2(c010, c110, wd);
        float2 c11v = LRP2(c011, c111, wd);
        float2 c0v  = LRP2(c00v, c10v, wh);
        float2 c1v  = LRP2(c01v, c11v, wh);
        float2 res  = LRP2(c0v,  c1v,  wf);
        #undef LRP2

        // ---- nearest label warp (RNE round matches jnp.round) ----
        int dn = (int)fminf(fmaxf(rintf(cd), 0.f), 127.f);
        int hn = (int)fminf(fmaxf(rintf(ch), 0.f), 159.f);
        int wn = (int)fminf(fmaxf(rintf(cw), 0.f), 159.f);
        float lv = lbl[((bD + dn) * Hh + hn) * Ww + wn];

        int po = ((bD + d) * Hh + hh) * Ww + ww;
        ((float2*)out_img)[po] = res;       // both channels, coalesced per half-wave
        out_lbl[po] = lv;
    }
}

extern "C" void kernel_launch(void* const* d_in, const int* in_sizes, int n_in,
                              void* d_out, int out_size, void* d_ws, size_t ws_size,
                              hipStream_t stream) {
    const float* image = (const float*)d_in[0];   // (2,128,160,160,2) f32
    const float* label = (const float*)d_in[1];   // (2,128,160,160,1) f32
    const float* cflow = (const float*)d_in[2];   // (2,4,4,4,3) f32 in [0,1)

    float* M       = (float*)d_ws;                // 1792 floats of fused weights
    float* out_img = (float*)d_out;               // image output first
    float* out_lbl = out_img + IMG_ELEMS;         // then label output

    build_tables_kernel<<<1, 448, 0, stream>>>(M);

    dim3 grid(Ww / 16, Hh / 16, (Bb * Dd) / 4);   // (10, 10, 64), 4 waves/block
    deform_kernel<<<grid, 128, 0, stream>>>(image, label, cflow, M, out_img, out_lbl);

    (void)in_sizes; (void)n_in; (void)out_size; (void)ws_size;
}